// GridBatchPooling_65309272703433
// MI455X (gfx1250) — compile-verified
//
#include <hip/hip_runtime.h>
#include <hip/hip_bf16.h>

// ---------------------------------------------------------------------------
// GridBatchPooling: segment-mean + representative scatter + gather + mask.
//   seg_sum[s,:]  = sum of x rows with ul_idx==s        (global f32 atomics, L2)
//   out[p,:]      = mask[p] * (ul_idx_inv[seg]==r ? seg_sum[seg,:]/max(cnt,1) : 0)
// Memory-bound (~270 MB total traffic -> ~12 us at 23.3 TB/s). No matmul
// structure -> WMMA not applicable; CDNA5 path used: async global->LDS
// double-buffered staging (ASYNCcnt) to overlap HBM reads with L2 atomics.
// ---------------------------------------------------------------------------

namespace {
constexpr int kB  = 4;
constexpr int kN  = 65536;
constexpr int kC  = 128;
constexpr int kS  = 32768;
constexpr int kBN = kB * kN;

constexpr int kThreads     = 256;                 // 8 waves (wave32)
constexpr int kTilePts     = 32;                  // points per LDS tile
constexpr int kTileFloats  = kTilePts * kC;       // 4096 floats = 16 KB
constexpr int kTileF4      = kTileFloats / 4;     // 1024 x b128
constexpr int kF4PerThread = kTileF4 / kThreads;  // 4 async b128 per thread
constexpr int kNumChunks   = kBN / kTilePts;      // 8192
constexpr int kAccumBlocks = 2048;

constexpr size_t kSegSumFloats   = (size_t)kS * kC;                    // 4 Mi floats
constexpr size_t kCntOffsetBytes = kSegSumFloats * sizeof(float);      // 16 MiB
constexpr size_t kWsBytes        = kCntOffsetBytes + (size_t)kS * sizeof(unsigned);
}  // namespace

typedef __attribute__((ext_vector_type(4))) int v4i;

__device__ __forceinline__ void async_load_b128(const float* gsrc, float* ldst) {
#if __has_builtin(__builtin_amdgcn_global_load_async_to_lds_b128)
  // Builtin signature (probe-confirmed): pointer params are 'v4i *' in the
  // generic address space. Order follows the llvm.amdgcn.*load*lds
  // convention: (global source, LDS destination, imm offset, imm cpol).
  __builtin_amdgcn_global_load_async_to_lds_b128(
      (v4i*)gsrc, (v4i*)ldst, /*offset=*/0, /*cpol=*/0);
#else
  unsigned loff =
      (unsigned)(unsigned long long)(__attribute__((address_space(3))) float*)ldst;
  asm volatile("global_load_async_to_lds_b128 %0, %1, off"
               :
               : "v"(loff), "v"(gsrc)
               : "memory");
#endif
}

__device__ __forceinline__ void wait_async_le4() {
#if __has_builtin(__builtin_amdgcn_s_wait_asynccnt)
  __builtin_amdgcn_s_wait_asynccnt(4);
#else
  asm volatile("s_wait_asynccnt 0x4" ::: "memory");
#endif
}

__device__ __forceinline__ void wait_async_le0() {
#if __has_builtin(__builtin_amdgcn_s_wait_asynccnt)
  __builtin_amdgcn_s_wait_asynccnt(0);
#else
  asm volatile("s_wait_asynccnt 0x0" ::: "memory");
#endif
}

// Pass 1: double-buffered async-staged segment accumulation.
__global__ __launch_bounds__(kThreads) void seg_accumulate_kernel(
    const float* __restrict__ x, const int* __restrict__ ul_idx,
    float* __restrict__ seg_sum, unsigned* __restrict__ seg_cnt) {
  __shared__ float tile[2][kTileFloats];  // 32 KB, double buffered

  const int t    = threadIdx.x;
  const int wave = t >> 5;
  const int lane = t & 31;

  int chunk = blockIdx.x;
  int cur   = 0;

  // Prologue: kick off first tile.
  if (chunk < kNumChunks) {
    const float* g = x + (size_t)chunk * kTileFloats;
    for (int i = 0; i < kF4PerThread; ++i) {
      const int f4 = t + i * kThreads;
      async_load_b128(g + f4 * 4, &tile[0][f4 * 4]);
    }
  }

  for (; chunk < kNumChunks; chunk += kAccumBlocks) {
    const int  next     = chunk + kAccumBlocks;
    const bool has_next = next < kNumChunks;

    if (has_next) {
      // Prefetch next tile into the other buffer while current is in flight.
      const float* g = x + (size_t)next * kTileFloats;
      for (int i = 0; i < kF4PerThread; ++i) {
        const int f4 = t + i * kThreads;
        async_load_b128(g + f4 * 4, &tile[cur ^ 1][f4 * 4]);
      }
      wait_async_le4();  // my 4 current-tile copies complete (issue order)
    } else {
      wait_async_le0();
    }
    __syncthreads();  // all waves' current-tile copies landed in LDS

    // Consume current tile: wave w owns points w*4..w*4+3, lane owns 4 channels.
    const int base_pt = chunk * kTilePts;
    for (int q = 0; q < 4; ++q) {
      const int    p   = wave * 4 + q;
      const int    seg = ul_idx[base_pt + p];
      const float4 v   = *(const float4*)&tile[cur][p * kC + lane * 4];
      float*       dst = seg_sum + (size_t)seg * kC + lane * 4;
      atomicAdd(dst + 0, v.x);
      atomicAdd(dst + 1, v.y);
      atomicAdd(dst + 2, v.z);
      atomicAdd(dst + 3, v.w);
      if (lane == 0) atomicAdd(&seg_cnt[seg], 1u);
    }
    __syncthreads();  // buffer 'cur' free for the next iteration's prefetch
    cur ^= 1;
  }
}

// Pass 2: mean + representative-slot gather + mask. One wave per point.
__global__ __launch_bounds__(kThreads) void finalize_gather_kernel(
    const float* __restrict__ seg_sum, const unsigned* __restrict__ seg_cnt,
    const int* __restrict__ idx, const float* __restrict__ mask,
    const int* __restrict__ ul_idx, const int* __restrict__ ul_idx_inv,
    float* __restrict__ out) {
  const int tid  = blockIdx.x * kThreads + threadIdx.x;
  const int p    = tid >> 5;   // point index
  const int lane = tid & 31;   // 4 channels per lane
  if (p >= kBN) return;

  const int       i0 = idx[2 * p + 0];
  const int       i1 = idx[2 * p + 1];
  const long long r  = (long long)i0 * kN + i1;

  float4      v = make_float4(0.f, 0.f, 0.f, 0.f);
  const float m = mask[p];

  if (r >= 0 && r < kBN) {
    const int seg = ul_idx[r];
    if (ul_idx_inv[seg] == (int)r) {  // gather hits the representative slot
      const float  n = fmaxf((float)seg_cnt[seg], 1.0f);
      const float4 s = *(const float4*)&seg_sum[(size_t)seg * kC + lane * 4];
      v.x = s.x / n;
      v.y = s.y / n;
      v.z = s.z / n;
      v.w = s.w / n;
    }
  }
  v.x *= m;
  v.y *= m;
  v.z *= m;
  v.w *= m;
  *(float4*)&out[(size_t)p * kC + lane * 4] = v;
}

extern "C" void kernel_launch(void* const* d_in, const int* in_sizes, int n_in,
                              void* d_out, int out_size, void* d_ws, size_t ws_size,
                              hipStream_t stream) {
  (void)in_sizes; (void)n_in; (void)out_size; (void)ws_size;

  const float* x          = (const float*)d_in[0];  // [B,N,C] f32
  const int*   idx        = (const int*)d_in[1];    // [B,N,2] i32
  const float* mask       = (const float*)d_in[2];  // [B,N,1] f32
  const int*   ul_idx     = (const int*)d_in[3];    // [BN]    i32
  const int*   ul_idx_inv = (const int*)d_in[4];    // [S]     i32
  float*       out        = (float*)d_out;          // [B,N,C] f32

  float*    seg_sum = (float*)d_ws;
  unsigned* seg_cnt = (unsigned*)((char*)d_ws + kCntOffsetBytes);

  // Workspace must be zeroed every call (we accumulate into it).
  (void)hipMemsetAsync(d_ws, 0, kWsBytes, stream);

  hipLaunchKernelGGL(seg_accumulate_kernel, dim3(kAccumBlocks), dim3(kThreads),
                     0, stream, x, ul_idx, seg_sum, seg_cnt);

  const int totalThreads = kBN * (kC / 4);  // one thread per (point, float4)
  hipLaunchKernelGGL(finalize_gather_kernel, dim3(totalThreads / kThreads),
                     dim3(kThreads), 0, stream, seg_sum, seg_cnt, idx, mask,
                     ul_idx, ul_idx_inv, out);
}